// LengthRegulator_43104291783440
// MI455X (gfx1250) — compile-verified
//
#include <hip/hip_runtime.h>
#include <math.h>

typedef __attribute__((ext_vector_type(2))) float v2f;
typedef __attribute__((ext_vector_type(4))) float v4f;
typedef __attribute__((ext_vector_type(8))) float v8f;

#define B_ 8
#define T_ 1024
#define D_ 384

// ---------------------------------------------------------------------------
// Kernel 1: total_dur[t] = sum_b duration_target[b][t]; exclusive scan -> offs
// Single block, 1024 threads, Hillis-Steele scan in LDS (10 steps).
// ---------------------------------------------------------------------------
__global__ void lr_scan_kernel(const int* __restrict__ dt,
                               int* __restrict__ dur,
                               int* __restrict__ offs) {
    __shared__ int s[T_];
    const int t = threadIdx.x;
    int sum = 0;
#pragma unroll
    for (int bb = 0; bb < B_; ++bb) sum += dt[bb * T_ + t];
    dur[t] = sum;
    s[t]   = sum;
    __syncthreads();
#pragma unroll
    for (int d = 1; d < T_; d <<= 1) {
        int v = (t >= d) ? s[t - d] : 0;
        __syncthreads();
        s[t] += v;
        __syncthreads();
    }
    offs[t] = s[t] - sum;   // exclusive prefix sum
}

// ---------------------------------------------------------------------------
// Kernel 2: duration_pred[b,t] = exp(dot(x[b,t,:], w) + bias) - 1
// One wave per 16-row tile, V_WMMA_F32_16X16X4_F32 accumulating over K=384.
// A layout (ISA 16x4 f32): lane<16 -> M=lane, K={0,1} in vgpr{0,1};
//                          lane>=16 -> M=lane-16, K={2,3}.
// B (4x16): vgpr r, lane-half h holds row K = r + 2*h. We replicate w[k+K]
// across all N columns, so every column of D holds the same dot product
// (N-striping independent). Row m result: lane (m<8?0:16), c[m%8].
// ---------------------------------------------------------------------------
__global__ void lr_pred_wmma_kernel(const float* __restrict__ x,
                                    const float* __restrict__ w,
                                    const float* __restrict__ bias,
                                    float* __restrict__ pred) {
    const int lane = threadIdx.x & 31;
    const int wave = threadIdx.x >> 5;
    const int tile = blockIdx.x * 8 + wave;        // 512 tiles of 16 rows
    const int half = lane >> 4;                    // 0: K base 0, 1: K base 2
    const int mrow = lane & 15;

    const float* xr = x + (size_t)(tile * 16 + mrow) * D_ + 2 * half;
    const float* wr = w + 2 * half;

    v8f c = {0.f, 0.f, 0.f, 0.f, 0.f, 0.f, 0.f, 0.f};
    for (int k = 0; k < D_; k += 4) {
        v2f a, bm;
        a.x  = xr[k];     a.y  = xr[k + 1];        // A: x row, K = k+2h, k+2h+1
        bm.x = wr[k];     bm.y = wr[k + 1];        // B: w replicated over N
        c = __builtin_amdgcn_wmma_f32_16x16x4_f32(
                /*neg_a=*/false, a, /*neg_b=*/false, bm,
                /*c_mod=*/(short)0, c, /*reuse_a=*/false, /*reuse_b=*/false);
    }

    const float bv = bias[0];
    if ((lane & 15) == 0) {                        // lanes 0 (rows 0-7) and 16 (rows 8-15)
        const int base = tile * 16 + 8 * half;
#pragma unroll
        for (int i = 0; i < 8; ++i)
            pred[base + i] = expf(c[i] + bv) - 1.0f;
    }
}

// ---------------------------------------------------------------------------
// Kernel 3: expanded[b, offs[t]+r, :] = x[b, t, :]  for r in [0, dur[t])
// grid = (T, B), 96 threads: each thread keeps one float4 of the source row
// in a register and streams dur[t] coalesced non-temporal B128 stores.
// Output is write-once (~350 MB) -> NT keeps it from evicting x out of L2.
// ---------------------------------------------------------------------------
__global__ void lr_expand_kernel(const float* __restrict__ x,
                                 const int* __restrict__ dur,
                                 const int* __restrict__ offs,
                                 float* __restrict__ out, int L) {
    const int t  = blockIdx.x;
    const int b  = blockIdx.y;
    const int q  = threadIdx.x;                    // 0..95 (float4 lane of row)
    const int n  = dur[t];
    if (n == 0) return;
    const int off = offs[t];

    const v4f v = ((const v4f*)(x + (size_t)(b * T_ + t) * D_))[q];
    v4f* dst = (v4f*)out + ((size_t)b * L + off) * (D_ / 4) + q;
    for (int r = 0; r < n; ++r) {
        __builtin_nontemporal_store(v, dst);
        dst += D_ / 4;
    }
}

// ---------------------------------------------------------------------------
extern "C" void kernel_launch(void* const* d_in, const int* in_sizes, int n_in,
                              void* d_out, int out_size, void* d_ws, size_t ws_size,
                              hipStream_t stream) {
    const float* x    = (const float*)d_in[0];   // [8,1024,384] f32
    const int*   dt   = (const int*)d_in[1];     // [8,1024] i32
    const float* w    = (const float*)d_in[2];   // [1,384] f32
    const float* bias = (const float*)d_in[3];   // [1] f32

    float* out = (float*)d_out;
    // out = concat(expanded [B,L,D], duration_pred [B,T]) flat
    const int L = (out_size - B_ * T_) / (B_ * D_);

    int* dur  = (int*)d_ws;          // [1024]
    int* offs = dur + T_;            // [1024]
    float* pred = out + (size_t)B_ * L * D_;

    hipLaunchKernelGGL(lr_scan_kernel, dim3(1), dim3(T_), 0, stream, dt, dur, offs);
    hipLaunchKernelGGL(lr_pred_wmma_kernel, dim3((B_ * T_ / 16) / 8), dim3(256), 0, stream,
                       x, w, bias, pred);
    hipLaunchKernelGGL(lr_expand_kernel, dim3(T_, B_), dim3(96), 0, stream,
                       x, dur, offs, out, L);
}